// GCN_77790447665815
// MI455X (gfx1250) — compile-verified
//
#include <hip/hip_runtime.h>
#include <hip/hip_bf16.h>
#include <math.h>

typedef float v2f __attribute__((ext_vector_type(2)));
typedef float v4f __attribute__((ext_vector_type(4)));
typedef float v8f __attribute__((ext_vector_type(8)));

#define HDIM 64
#define GNUM 256

// ---------------------------------------------------------------------------
// Guaranteed-native fire-and-forget f32 atomic add at device scope
// (L2-side atomic unit; no CAS loop, no return, STOREcnt-tracked).
// ---------------------------------------------------------------------------
__device__ __forceinline__ void gl_atomic_add_f32(float* p, float v) {
    asm volatile("global_atomic_add_f32 %0, %1, off scope:SCOPE_DEV"
                 :: "v"(p), "v"(v) : "memory");
}

// ---------------------------------------------------------------------------
// Async memory->LDS staging (CDNA5 GLOBAL_LOAD_ASYNC_TO_LDS_B128, ASYNCcnt).
// Probe round 2 showed the builtin exists with signature:
//   (v4i global*, v4i lds*, int offset, int cpol)
// ---------------------------------------------------------------------------
#if __has_builtin(__builtin_amdgcn_global_load_async_to_lds_b128)
#define HAVE_ASYNC_LDS 1
typedef int v4i __attribute__((vector_size(16)));
typedef __attribute__((address_space(1))) v4i* as1_v4i_ptr;
typedef __attribute__((address_space(3))) v4i* as3_v4i_ptr;
__device__ __forceinline__ void async_copy16(const void* g, void* lds) {
    __builtin_amdgcn_global_load_async_to_lds_b128(
        (as1_v4i_ptr)g, (as3_v4i_ptr)lds, 0, 0);
}
__device__ __forceinline__ void async_wait0() {
#if __has_builtin(__builtin_amdgcn_s_wait_asynccnt)
    __builtin_amdgcn_s_wait_asynccnt(0);
#else
    asm volatile("s_wait_asynccnt 0" ::: "memory");
#endif
}
#else
#define HAVE_ASYNC_LDS 0
#endif

// ---------------------------------------------------------------------------
// 1) degree count over col (self loops added analytically later)
// ---------------------------------------------------------------------------
__global__ void k_deg(const int* __restrict__ col, float* __restrict__ deg, int E) {
    int e = blockIdx.x * blockDim.x + threadIdx.x;
    if (e < E) gl_atomic_add_f32(&deg[col[e]], 1.0f);
}

// dinv[i] = rsqrt(deg[i] + 1)   (in place: deg buffer becomes dinv)
__global__ void k_dinv(float* __restrict__ d, int N) {
    int i = blockIdx.x * blockDim.x + threadIdx.x;
    if (i < N) d[i] = rsqrtf(d[i] + 1.0f);
}

// ---------------------------------------------------------------------------
// 2) fp32 WMMA GEMM:  C[N x 64] = A[N x K] @ W[K x 64]
//    block = 256 threads = 8 wave32; block tile = 32(M) x 64(N)
//    wave (w) -> mtile = (w>>2)*16, ntile = (w&3)*16 ; K-loop in steps of 4
//    V_WMMA_F32_16X16X4_F32: full fp32 (workload is memory bound, keep exact)
//    A/W tiles staged via async global->LDS DMA when available.
// ---------------------------------------------------------------------------
template <int K>
__global__ __launch_bounds__(256) void k_gemm_wmma(const float* __restrict__ A,
                                                   const float* __restrict__ W,
                                                   float* __restrict__ C, int Nrows) {
    constexpr int PITCH = K + 4;              // pad to avoid K-stride bank conflicts
    __shared__ float sW[K * HDIM];
    __shared__ float sA[32 * PITCH];

    const int tid  = threadIdx.x;
    const int row0 = blockIdx.x * 32;

#if HAVE_ASYNC_LDS
    // stage W (row-major [K][64], contiguous) via async DMA
    for (int t = tid; t < K * (HDIM / 4); t += 256)
        async_copy16((const v4f*)W + t, (v4f*)sW + t);
    // stage 32 x K tile of A via async DMA (zero-fill out-of-range rows)
    for (int t = tid; t < 32 * (K / 4); t += 256) {
        int r  = t / (K / 4);
        int kk = (t % (K / 4)) * 4;
        int gr = row0 + r;
        if (gr < Nrows)
            async_copy16(&A[(size_t)gr * K + kk], &sA[r * PITCH + kk]);
        else
            *(v4f*)&sA[r * PITCH + kk] = v4f{};
    }
    async_wait0();                 // drain this wave's ASYNCcnt before barrier
#else
    for (int t = tid; t < K * (HDIM / 4); t += 256)
        ((v4f*)sW)[t] = ((const v4f*)W)[t];
    for (int t = tid; t < 32 * (K / 4); t += 256) {
        int r  = t / (K / 4);
        int kk = (t % (K / 4)) * 4;
        int gr = row0 + r;
        v4f v = {};
        if (gr < Nrows) v = *(const v4f*)&A[(size_t)gr * K + kk];
        *(v4f*)&sA[r * PITCH + kk] = v;
    }
#endif
    __syncthreads();

    const int wid   = tid >> 5;
    const int lane  = tid & 31;
    const int mtile = (wid >> 2) * 16;
    const int ntile = (wid & 3) * 16;
    const int mn    = lane & 15;          // M index for A frag, N index for B/C frags
    const int khalf = (lane >> 4) * 2;    // lanes 16-31 hold K+2,K+3

    v8f acc = {};
#pragma unroll
    for (int kb = 0; kb < K; kb += 4) {
        // A 16x4 frag: lane m holds A[m][kb+khalf+0], A[m][kb+khalf+1]
        v2f a = *(const v2f*)&sA[(mtile + mn) * PITCH + kb + khalf];
        // B 4x16 frag (mirror layout): lane n holds W[kb+khalf+0][n], W[kb+khalf+1][n]
        v2f b;
        b.x = sW[(kb + khalf) * HDIM + ntile + mn];
        b.y = sW[(kb + khalf + 1) * HDIM + ntile + mn];
        acc = __builtin_amdgcn_wmma_f32_16x16x4_f32(false, a, false, b,
                                                    (short)0, acc, false, false);
    }

    // C/D layout: VGPR r -> M = r + 8*(lane>=16), N = lane&15
    const int hi = lane >> 4;
#pragma unroll
    for (int r = 0; r < 8; ++r) {
        int gm = row0 + mtile + r + hi * 8;
        if (gm < Nrows) C[(size_t)gm * HDIM + ntile + mn] = acc[r];
    }
}

// ---------------------------------------------------------------------------
// 3) edge scatter-add: agg[col] += dinv[row]*dinv[col] * hw[row]
//    16 lanes per edge, float4 each -> coalesced 256B gather per edge,
//    native L2 f32 atomics for the scatter (the dominant cost of the model)
// ---------------------------------------------------------------------------
__global__ void k_edge_agg(const int* __restrict__ row, const int* __restrict__ col,
                           const float* __restrict__ dinv, const float* __restrict__ hw,
                           float* __restrict__ agg, int E) {
    int t = blockIdx.x * blockDim.x + threadIdx.x;
    int e = t >> 4;
    int q = (t & 15) * 4;
    if (e >= E) return;
    int r = row[e], c = col[e];
    float nrm = dinv[r] * dinv[c];
    v4f v = *(const v4f*)&hw[(size_t)r * HDIM + q];
    float* dst = &agg[(size_t)c * HDIM + q];
    gl_atomic_add_f32(dst + 0, nrm * v.x);
    gl_atomic_add_f32(dst + 1, nrm * v.y);
    gl_atomic_add_f32(dst + 2, nrm * v.z);
    gl_atomic_add_f32(dst + 3, nrm * v.w);
}

// self-loop term + bias (+ optional ReLU), in place in agg
__global__ void k_finalize(const float* __restrict__ hw, const float* __restrict__ dinv,
                           const float* __restrict__ bias, float* __restrict__ agg,
                           int relu, int N) {
    int t = blockIdx.x * blockDim.x + threadIdx.x;
    if (t >= N * HDIM) return;
    int i = t >> 6, f = t & 63;
    float d = dinv[i];
    float v = agg[t] + d * d * hw[t] + bias[f];
    agg[t] = relu ? fmaxf(v, 0.0f) : v;
}

// ---------------------------------------------------------------------------
// 4) per-graph pooling (mean+max via binary search on sorted batch) + MLP head
// ---------------------------------------------------------------------------
__global__ __launch_bounds__(64) void k_pool_mlp(const float* __restrict__ h,
                                                 const int* __restrict__ batch,
                                                 const float* __restrict__ fcW1,
                                                 const float* __restrict__ fcb1,
                                                 const float* __restrict__ fcW2,
                                                 const float* __restrict__ fcb2,
                                                 float* __restrict__ out, int N) {
    __shared__ float p[2 * HDIM];
    __shared__ float z[HDIM];
    __shared__ float red[HDIM];
    __shared__ int seg[2];

    int g = blockIdx.x, f = threadIdx.x;
    if (f < 2) {  // lower_bound(batch, g) and lower_bound(batch, g+1)
        int target = g + f, lo = 0, hi = N;
        while (lo < hi) {
            int mid = (lo + hi) >> 1;
            if (batch[mid] < target) lo = mid + 1; else hi = mid;
        }
        seg[f] = lo;
    }
    __syncthreads();
    int s = seg[0], e = seg[1];

    float sum = 0.0f, mx = -INFINITY;
    for (int i = s; i < e; ++i) {
        float v = h[(size_t)i * HDIM + f];
        sum += v;
        mx = fmaxf(mx, v);
    }
    int cnt = e - s;
    p[f]        = sum / (float)(cnt > 0 ? cnt : 1);
    p[HDIM + f] = (cnt > 0) ? mx : 0.0f;   // isfinite guard for empty segments
    __syncthreads();

    float zz = fcb1[f];
    for (int k = 0; k < 2 * HDIM; ++k) zz += p[k] * fcW1[k * HDIM + f];
    z[f] = fmaxf(zz, 0.0f);
    __syncthreads();

    red[f] = z[f] * fcW2[f];
    __syncthreads();
    for (int st = 32; st > 0; st >>= 1) {
        if (f < st) red[f] += red[f + st];
        __syncthreads();
    }
    if (f == 0) out[g] = red[0] + fcb2[0];
}

// ---------------------------------------------------------------------------
extern "C" void kernel_launch(void* const* d_in, const int* in_sizes, int n_in,
                              void* d_out, int out_size, void* d_ws, size_t ws_size,
                              hipStream_t stream) {
    const float* x    = (const float*)d_in[0];
    const int*   ei   = (const int*)d_in[1];
    const int*   bat  = (const int*)d_in[2];
    const float* W1   = (const float*)d_in[3];
    const float* b1   = (const float*)d_in[4];
    const float* W2   = (const float*)d_in[5];
    const float* b2   = (const float*)d_in[6];
    const float* fcW1 = (const float*)d_in[7];
    const float* fcb1 = (const float*)d_in[8];
    const float* fcW2 = (const float*)d_in[9];
    const float* fcb2 = (const float*)d_in[10];
    float* out = (float*)d_out;

    const int E = in_sizes[1] / 2;
    const int N = in_sizes[2];
    const int* row = ei;
    const int* col = ei + E;

    // workspace: dinv[N] | hw[N*64] | agg[N*64]
    char* ws = (char*)d_ws;
    size_t ofs = 0;
    float* dinv = (float*)(ws + ofs); ofs += ((size_t)N * 4 + 255) & ~(size_t)255;
    float* hw   = (float*)(ws + ofs); ofs += ((size_t)N * HDIM * 4 + 255) & ~(size_t)255;
    float* agg  = (float*)(ws + ofs);

    const int T = 256;
    const int gemmGrid = (N + 31) / 32;
    const int edgeGrid = (int)(((long long)E * 16 + T - 1) / T);
    const int featGrid = (int)(((long long)N * HDIM + T - 1) / T);

    // degree / norm
    (void)hipMemsetAsync(dinv, 0, (size_t)N * 4, stream);
    (void)hipMemsetAsync(agg, 0, (size_t)N * HDIM * 4, stream);
    k_deg<<<(E + T - 1) / T, T, 0, stream>>>(col, dinv, E);
    k_dinv<<<(N + T - 1) / T, T, 0, stream>>>(dinv, N);

    // layer 1: xw = x@W1 ; aggregate ; +selfloop +bias +ReLU
    k_gemm_wmma<128><<<gemmGrid, T, 0, stream>>>(x, W1, hw, N);
    k_edge_agg<<<edgeGrid, T, 0, stream>>>(row, col, dinv, hw, agg, E);
    k_finalize<<<featGrid, T, 0, stream>>>(hw, dinv, b1, agg, 1, N);   // agg -> h1

    // layer 2: hw = h1@W2 ; aggregate ; +selfloop +bias
    k_gemm_wmma<64><<<gemmGrid, T, 0, stream>>>(agg, W2, hw, N);
    (void)hipMemsetAsync(agg, 0, (size_t)N * HDIM * 4, stream);        // h1 consumed
    k_edge_agg<<<edgeGrid, T, 0, stream>>>(row, col, dinv, hw, agg, E);
    k_finalize<<<featGrid, T, 0, stream>>>(hw, dinv, b2, agg, 0, N);   // agg -> h2

    // pooling + MLP head
    k_pool_mlp<<<GNUM, 64, 0, stream>>>(agg, bat, fcW1, fcb1, fcW2, fcb2, out, N);
}